// ProcessMask_81716047773715
// MI455X (gfx1250) — compile-verified
//
#include <hip/hip_runtime.h>
#include <hip/hip_bf16.h>
#include <math.h>

typedef __attribute__((ext_vector_type(2))) float v2f;
typedef __attribute__((ext_vector_type(8))) float v8f;

#define BATCH   8
#define A_DIM   8400
#define C_DIM   32
#define HP      160
#define WP      160
#define NMASK   100
#define NPIX    (HP * WP)        // 25600
#define ROWTILES (NPIX / 16)     // 1600
#define NTILES_N 7               // ceil(100/16)
// box scale: 160/640 = 0.25 in both x and y
#define BSCALE  0.25f

__global__ __launch_bounds__(256)
void process_mask_wmma(const float* __restrict__ masks,   // (B, A, C)
                       const float* __restrict__ protos,  // (B, HP, WP, C)
                       const float* __restrict__ boxes,   // (B, N, 4)
                       const int*   __restrict__ idx,     // (B, N)
                       float*       __restrict__ out)     // (B, HP, WP, N)
{
    const int lane        = threadIdx.x & 31;
    const int waveInBlock = threadIdx.x >> 5;
    const int waveGlobal  = blockIdx.x * 8 + waveInBlock;   // 0 .. 12799
    const int rowTile     = waveGlobal % ROWTILES;
    const int b           = waveGlobal / ROWTILES;

    const int g    = lane >> 4;     // lane-half select (0: lanes 0-15, 1: lanes 16-31)
    const int mRow = lane & 15;     // A-matrix row / B,C,D column within tile

    // A 16-pixel tile never crosses an image row (160 % 16 == 0):
    const int   pixBase = rowTile * 16;
    const float hRow    = (float)(pixBase / WP);   // wave-uniform
    const int   wBase   = pixBase % WP;

    // ---- Load A (protos tile, 16 pixels x 32 coeffs) ONCE, kept in regs ----
    // f32 16x4 A layout: lane<16 -> {K=0,K=1}; lane>=16 -> {K=2,K=3} per chunk.
    const float* aRow = protos + ((size_t)b * NPIX + (pixBase + mRow)) * C_DIM;
    v2f a[8];
#pragma unroll
    for (int kc = 0; kc < 8; ++kc) {
        a[kc] = *(const v2f*)(aRow + kc * 4 + g * 2);
    }

    // ---- Loop over 7 tiles of 16 mask columns ----
    for (int nt = 0; nt < NTILES_N; ++nt) {
        const int n     = nt * 16 + mRow;       // this lane's mask column
        const bool nval = (n < NMASK);

        // Gather B (pred coeffs): row = masks[b, idx[b,n], :]; zeros if n OOB.
        v2f bb[8];
        if (nval) {
            const int row = idx[b * NMASK + n];
            const float* bRow = masks + ((size_t)b * A_DIM + row) * C_DIM;
#pragma unroll
            for (int kc = 0; kc < 8; ++kc) {
                bb[kc] = *(const v2f*)(bRow + kc * 4 + g * 2);
            }
        } else {
#pragma unroll
            for (int kc = 0; kc < 8; ++kc) bb[kc] = (v2f)(0.0f);
        }

        // Box for this lane's mask column (scaled into mask-grid coords).
        float bx1 = 0.f, by1 = 0.f, bx2 = 0.f, by2 = 0.f;
        if (nval) {
            const float4 bx = *(const float4*)(boxes + ((size_t)b * NMASK + n) * 4);
            bx1 = bx.x * BSCALE;
            by1 = bx.y * BSCALE;
            bx2 = bx.z * BSCALE;
            by2 = bx.w * BSCALE;
        }
        // h-part of the crop predicate is per-lane but r-invariant: hoist it.
        const bool hIn = (hRow >= by1) & (hRow < by2);

        // ---- 8 chained fp32 WMMAs: D = A(16x4) * B(4x16) + C, K = 32 ----
        v8f acc = {};
#pragma unroll
        for (int kc = 0; kc < 8; ++kc) {
            acc = __builtin_amdgcn_wmma_f32_16x16x4_f32(
                /*neg_a=*/false, a[kc],
                /*neg_b=*/false, bb[kc],
                /*c_mod=*/(short)0, acc,
                /*reuse_a=*/false, /*reuse_b=*/false);
        }

        // ---- Epilogue: sigmoid -> crop -> threshold -> store ----
        // C/D layout: VGPR r, lane half g -> pixel row (r + 8*g); column = mRow.
        // sigmoid(x) > 0.5  <=>  x > 0  (exact), so gate on raw accumulator and
        // use hardware v_rcp_f32 (1 ulp) for the stored value.
        if (nval) {
#pragma unroll
            for (int r = 0; r < 8; ++r) {
                const int   pix = pixBase + r + 8 * g;
                const float w   = (float)(wBase + r + 8 * g);
                const float x   = acc[r];
                const float val = __builtin_amdgcn_rcpf(1.0f + __expf(-x));
                const bool  keep = hIn & (w >= bx1) & (w < bx2) & (x > 0.0f);
                out[((size_t)b * NPIX + pix) * NMASK + n] = keep ? val : 0.0f;
            }
        }
    }
}

extern "C" void kernel_launch(void* const* d_in, const int* in_sizes, int n_in,
                              void* d_out, int out_size, void* d_ws, size_t ws_size,
                              hipStream_t stream) {
    const float* masks  = (const float*)d_in[0];  // (8, 8400, 32)
    const float* protos = (const float*)d_in[1];  // (8, 160, 160, 32)
    const float* boxes  = (const float*)d_in[2];  // (8, 100, 4)
    const int*   idx    = (const int*)d_in[3];    // (8, 100)
    float*       out    = (float*)d_out;          // (8, 160, 160, 100)

    // 8 batches * 1600 row-tiles = 12800 waves; 8 waves per 256-thread block.
    const int blocks = (BATCH * ROWTILES) / 8;    // 1600
    process_mask_wmma<<<dim3(blocks), dim3(256), 0, stream>>>(
        masks, protos, boxes, idx, out);
}